// LinearAttentionKNO1D_28896539968127
// MI455X (gfx1250) — compile-verified
//
#include <hip/hip_runtime.h>
#include <hip/hip_bf16.h>
#include <math.h>
#include <stdint.h>

// ---------------- problem constants (match reference) ----------------
constexpr int B  = 2;
constexpr int C  = 256;
constexpr int N  = 32768;
constexpr int H  = 32;
constexpr int M  = 16;
constexpr int T  = 256;
constexpr int THID = 64;   // kernel_time_hidden
constexpr int DH = 8;      // head dim
constexpr int DK = 32;     // 2*M

typedef float v2f __attribute__((ext_vector_type(2)));
typedef float v8f __attribute__((ext_vector_type(8)));

#if defined(__HIP_DEVICE_COMPILE__) && __has_builtin(__builtin_amdgcn_wmma_f32_16x16x4_f32)
#define USE_WMMA_F32 1
#else
#define USE_WMMA_F32 0
#endif

__device__ __forceinline__ v8f wmma_f32_16x16x4(v2f a, v2f b, v8f c) {
#if USE_WMMA_F32
    // 8 args: (neg_a, A, neg_b, B, c_mod, C, reuse_a, reuse_b)
    return __builtin_amdgcn_wmma_f32_16x16x4_f32(false, a, false, b, (short)0, c,
                                                 false, false);
#else
    c[0] += a[0] * b[0] + a[1] * b[1];
    return c;
#endif
}

// CDNA5 async global->LDS copy (16B per lane), tracked by ASYNCcnt.
__device__ __forceinline__ void async_copy_b128(const void* lds_dst, const void* gsrc) {
#if defined(__HIP_DEVICE_COMPILE__)
    uint32_t lds = (uint32_t)(uintptr_t)lds_dst;   // flat->LDS uses addr[31:0]
    uint64_t g   = (uint64_t)(uintptr_t)gsrc;
    asm volatile("global_load_async_to_lds_b128 %0, %1, off"
                 :: "v"(lds), "v"(g) : "memory");
#else
    (void)lds_dst; (void)gsrc;
#endif
}

__device__ __forceinline__ void wait_async_all() {
#if defined(__HIP_DEVICE_COMPILE__)
    asm volatile("s_wait_asynccnt 0x0" ::: "memory");
#endif
}

__device__ __forceinline__ float silu(float x) { return x / (1.0f + __expf(-x)); }

// ---------------------------------------------------------------------
// Kernel A: time conditioning (shiftV and per-head lengthscale ell).
// ---------------------------------------------------------------------
__global__ void prep_kernel(const float* __restrict__ temb,
                            const float* __restrict__ tproj_w,
                            const float* __restrict__ tproj_b,
                            const float* __restrict__ vproj_w,
                            const float* __restrict__ vproj_b,
                            const float* __restrict__ tw1,
                            const float* __restrict__ tb1,
                            const float* __restrict__ tw2,
                            const float* __restrict__ tb2,
                            const float* __restrict__ log_ell,
                            float* __restrict__ shiftV,
                            float* __restrict__ ellb) {
    __shared__ float st[B][T];     // silu(temb)
    __shared__ float sh[B][C];     // shift
    __shared__ float hm[B][THID];  // hidden MLP activations
    const int t = threadIdx.x;

    for (int b = 0; b < B; ++b) st[b][t] = silu(temb[b * T + t]);
    if (t < B * THID) {  // hmid (raw temb input)
        int b = t / THID, j = t % THID;
        float s = tb1[j];
        for (int k = 0; k < T; ++k) s = fmaf(temb[b * T + k], tw1[j * T + k], s);
        hm[b][j] = silu(s);
    }
    __syncthreads();
    for (int b = 0; b < B; ++b) {  // shift[b][c], c = t
        float s = tproj_b[t];
        for (int k = 0; k < T; ++k) s = fmaf(st[b][k], tproj_w[t * T + k], s);
        sh[b][t] = s;
    }
    __syncthreads();
    for (int b = 0; b < B; ++b) {  // shiftV[b][o], o = t
        float s = vproj_b[t];
        for (int c = 0; c < C; ++c) s = fmaf(vproj_w[t * C + c], sh[b][c], s);
        shiftV[b * C + t] = s;
    }
    if (t < B * H) {  // ell
        int b = t / H, h = t % H;
        float d = tb2[h];
        for (int j = 0; j < THID; ++j) d = fmaf(hm[b][j], tw2[h * THID + j], d);
        ellb[b * H + h] = __expf(log_ell[h] + d);
    }
}

// ---------------------------------------------------------------------
// Kernel B: trapezoid quadrature weights over sorted coords.
// ---------------------------------------------------------------------
__global__ void quad_kernel(const float* __restrict__ xc, float* __restrict__ wq) {
    int idx = blockIdx.x * blockDim.x + threadIdx.x;  // 0 .. B*N-1
    int b = idx / N, n = idx % N;
    const float* x = xc + (size_t)b * N;
    float left  = (n > 0)     ? x[n] - x[n - 1] : 0.0f;
    float right = (n < N - 1) ? x[n + 1] - x[n] : 0.0f;
    wq[idx] = 0.5f * (left + right);
}

// ---------------------------------------------------------------------
// Kernel C/E2: WMMA fp32 GEMM   OUT[b][o][n] = sum_c W[o][c] * IN[b][c][n]
// Block tile: 256(o) x 64(n); 8 waves; wave = 32 o-rows x 64 n (8 accums,
// two 16-row o-subtiles reuse every B fragment -> ~2x LDS arithmetic
// intensity). Both operands double-buffered via ASYNC global->LDS b128.
// Epilogue: (acc + rowAdd0[b][o] + rowAdd1[o]) * colScale[b][n]
//   mode 0: store (B,H,N,DH) layout (Vw);  mode 1: store (B,C,N) layout.
// ---------------------------------------------------------------------
__global__ void gemm_kernel(const float* __restrict__ W,
                            const float* __restrict__ IN,
                            float* __restrict__ OUT,
                            const float* __restrict__ rowAdd0, int rowAdd0BStride,
                            const float* __restrict__ rowAdd1,
                            const float* __restrict__ colScale,
                            int mode) {
    constexpr int NT  = 64;   // n tile
    constexpr int KC  = 16;   // K chunk in LDS
    constexpr int XSP = 68;   // xs row stride (16B-aligned rows, bank-spread)
    constexpr int WSP = 20;   // wsa row stride (16B-aligned rows, 16 distinct banks)
    __shared__ float xs[2][KC * XSP];   // B-operand chunks: 2 x 4.25 KB
    __shared__ float wsa[2][C * WSP];   // A-operand chunks: 2 x 20 KB

    const int b      = blockIdx.z;
    const int nBase  = blockIdx.x * NT;
    const int oBase  = blockIdx.y * 256;  // grid.y == 1 -> 0
    const int t      = threadIdx.x;
    const int wave   = t >> 5;
    const int lane   = t & 31;
    const int laneLo = lane & 15;
    const int laneHi = lane >> 4;

    const float* INb = IN + (size_t)b * C * N;

    // async-stage one K chunk (both operands) into LDS buffer `buf`
    auto stage = [&](int buf, int kb) {
        // xs: KC x NT = 256 float4, 1 per thread
        {
            int r  = t >> 4;               // 16 float4 per row
            int c4 = (t & 15) << 2;
            async_copy_b128(&xs[buf][r * XSP + c4],
                            &INb[(size_t)(kb + r) * N + nBase + c4]);
        }
        // wsa: 256 rows x KC = 1024 float4, 4 per thread
#pragma unroll
        for (int i = 0; i < 4; ++i) {
            int idx = i * 256 + t;
            int row = idx >> 2;            // 4 float4 per row
            int c4  = (idx & 3) << 2;
            async_copy_b128(&wsa[buf][row * WSP + c4],
                            &W[(size_t)(oBase + row) * C + kb + c4]);
        }
    };

    v8f acc[2][4] = {};

    stage(0, 0);
    wait_async_all();
    __syncthreads();

    for (int kb = 0, buf = 0; kb < C; kb += KC, buf ^= 1) {
        if (kb + KC < C) stage(buf ^ 1, kb + KC);  // prefetch next chunk
#pragma unroll
        for (int ks = 0; ks < KC; ks += 4) {
            const int kk = ks + laneHi * 2;
            v2f a0 = *(const v2f*)&wsa[buf][(wave * 32 + laneLo) * WSP + kk];
            v2f a1 = *(const v2f*)&wsa[buf][(wave * 32 + 16 + laneLo) * WSP + kk];
#pragma unroll
            for (int j = 0; j < 4; ++j) {
                v2f bf;
                bf.x = xs[buf][kk * XSP + j * 16 + laneLo];        // B[kk][col]
                bf.y = xs[buf][(kk + 1) * XSP + j * 16 + laneLo];  // B[kk+1][col]
                acc[0][j] = wmma_f32_16x16x4(a0, bf, acc[0][j]);
                acc[1][j] = wmma_f32_16x16x4(a1, bf, acc[1][j]);
            }
        }
        wait_async_all();   // my wave's prefetch landed in LDS
        __syncthreads();    // everyone done computing + staging
    }

    // epilogue: lane holds col n = nBase + j*16 + laneLo, rows o0..o0+7
#pragma unroll
    for (int s = 0; s < 2; ++s) {
        const int o0 = oBase + wave * 32 + s * 16 + laneHi * 8;
        float ra[8];
#pragma unroll
        for (int i = 0; i < 8; ++i) {
            float r = rowAdd0 ? rowAdd0[b * rowAdd0BStride + o0 + i] : 0.0f;
            if (rowAdd1) r += rowAdd1[o0 + i];
            ra[i] = r;
        }
#pragma unroll
        for (int j = 0; j < 4; ++j) {
            const int n = nBase + j * 16 + laneLo;
            const float sc = colScale ? colScale[(size_t)b * N + n] : 1.0f;
            float vals[8];
#pragma unroll
            for (int i = 0; i < 8; ++i) vals[i] = (acc[s][j][i] + ra[i]) * sc;
            if (mode == 0) {
                const int h = o0 >> 3;  // o0 is a multiple of 8 -> dh = i
                float* dst = OUT + (((size_t)b * H + h) * N + n) * DH;
                *(float4*)&dst[0] = *(float4*)&vals[0];
                *(float4*)&dst[4] = *(float4*)&vals[4];
            } else {
#pragma unroll
                for (int i = 0; i < 8; ++i)
                    OUT[((size_t)b * C + o0 + i) * N + n] = vals[i];
            }
        }
    }
}

// ---------------------------------------------------------------------
// Kernel D: context[b][h][k][d] = sum_n phi[b][h][n][k] * Vw[b][h][n][d]
// One block per (b,h); 256 threads each own one (k,d) scalar accumulator.
// ---------------------------------------------------------------------
__global__ void ctx_kernel(const float* __restrict__ xc,
                           const float* __restrict__ omega0,
                           const float* __restrict__ ellb,
                           const float* __restrict__ Vw,
                           float* __restrict__ ctxb) {
    const int b = blockIdx.x / H;
    const int h = blockIdx.x % H;
    const int t = threadIdx.x;

    __shared__ float omx[M];
    __shared__ float phs[64][DK];  // phi chunk: 64 n x 32 k
    __shared__ float vsF[64 * DH]; // V chunk:   64 n x 8 d

    const float inv_ell = 1.0f / ellb[b * H + h];
    if (t < M) omx[t] = omega0[h * M + t] * inv_ell;
    __syncthreads();

    const int nl = t & 63;       // n within chunk (phi producer role)
    const int kg = t >> 6;       // k group 0..3  (8 k's each)
    const int kT = t >> 3;       // k for accumulation role
    const int dT = t & 7;        // d for accumulation role
    const float* xcb = xc + (size_t)b * N;
    const float* vsrc = Vw + ((size_t)b * H + h) * N * DH;

    float ctxv = 0.0f;
    for (int base = 0; base < N; base += 64) {
        const float xcn = xcb[base + nl];
        __syncthreads();  // previous accumulation finished
#pragma unroll
        for (int u = 0; u < 8; ++u) {
            int k = kg * 8 + u;
            int m = k & 15;
            float s, c;
            __sincosf(xcn * omx[m], &s, &c);
            phs[nl][k] = 0.25f * ((k < 16) ? c : s);  // M^-0.5 = 0.25
        }
        // V chunk: 512 contiguous floats, float2 per thread
        *(float2*)&vsF[t * 2] = *(const float2*)&vsrc[(size_t)base * DH + t * 2];
        __syncthreads();
#pragma unroll 8
        for (int nn = 0; nn < 64; ++nn)
            ctxv = fmaf(phs[nn][kT], vsF[nn * DH + dT], ctxv);
    }
    ctxb[((size_t)b * H + h) * (DK * DH) + t] = ctxv;  // t == k*8+d
}

// ---------------------------------------------------------------------
// Kernel E1: attn[b][c=h*8+d][n] = sum_k phi[b][h][n][k] * context[b][h][k][d]
// ---------------------------------------------------------------------
__global__ void attn_kernel(const float* __restrict__ xc,
                            const float* __restrict__ omega0,
                            const float* __restrict__ ellb,
                            const float* __restrict__ ctxb,
                            float* __restrict__ attn) {
    const int b = blockIdx.z;
    const int h = blockIdx.y;
    const int t = threadIdx.x;
    const int n = blockIdx.x * 256 + t;

    __shared__ float cs[DK * DH];
    __shared__ float omx[M];
    cs[t] = ctxb[((size_t)b * H + h) * (DK * DH) + t];
    if (t < M) omx[t] = omega0[h * M + t] / ellb[b * H + h];
    __syncthreads();

    const float xcn = xc[(size_t)b * N + n];
    float p[DK];
#pragma unroll
    for (int m = 0; m < M; ++m) {
        float s, c;
        __sincosf(xcn * omx[m], &s, &c);
        p[m]     = 0.25f * c;
        p[m + M] = 0.25f * s;
    }
#pragma unroll
    for (int d = 0; d < DH; ++d) {
        float a = 0.0f;
#pragma unroll
        for (int k = 0; k < DK; ++k) a = fmaf(p[k], cs[k * DH + d], a);
        attn[((size_t)b * C + h * DH + d) * N + n] = a;
    }
}

// ---------------------------------------------------------------------
extern "C" void kernel_launch(void* const* d_in, const int* in_sizes, int n_in,
                              void* d_out, int out_size, void* d_ws, size_t ws_size,
                              hipStream_t stream) {
    const float* x         = (const float*)d_in[0];
    const float* xcoords   = (const float*)d_in[1];
    const float* temb      = (const float*)d_in[2];
    const float* tproj_w   = (const float*)d_in[3];
    const float* tproj_b   = (const float*)d_in[4];
    const float* vproj_w   = (const float*)d_in[5];
    const float* vproj_b   = (const float*)d_in[6];
    const float* outproj_w = (const float*)d_in[7];
    const float* outproj_b = (const float*)d_in[8];
    const float* out_bias  = (const float*)d_in[9];
    const float* omega0    = (const float*)d_in[10];
    const float* log_ell   = (const float*)d_in[11];
    const float* tw1       = (const float*)d_in[12];
    const float* tb1       = (const float*)d_in[13];
    const float* tw2       = (const float*)d_in[14];
    const float* tb2       = (const float*)d_in[15];
    float* out = (float*)d_out;

    // workspace layout (floats)
    float* wsf    = (float*)d_ws;
    float* shiftV = wsf;                 // B*C            = 512
    float* ellb   = wsf + 512;           // B*H            = 64
    float* wq     = wsf + 576;           // B*N            = 65536
    float* ctxb   = wsf + 66112;         // B*H*DK*DH      = 16384
    float* Vw     = wsf + 82496;         // B*C*N (64 MB); reused as attn after ctx

    prep_kernel<<<1, 256, 0, stream>>>(temb, tproj_w, tproj_b, vproj_w, vproj_b,
                                       tw1, tb1, tw2, tb2, log_ell, shiftV, ellb);
    quad_kernel<<<(B * N) / 256, 256, 0, stream>>>(xcoords, wq);

    // vproj: Vw = (vproj_w @ (x + shift) + vb) * w   (stored (B,H,N,DH))
    dim3 ggrid(N / 64, 1, B);
    gemm_kernel<<<ggrid, 256, 0, stream>>>(vproj_w, x, Vw,
                                           shiftV, C, nullptr, wq, /*mode=*/0);

    ctx_kernel<<<B * H, 256, 0, stream>>>(xcoords, omega0, ellb, Vw, ctxb);

    // attn overwrites the Vw region (Vw no longer needed)
    attn_kernel<<<dim3(N / 256, H, B), 256, 0, stream>>>(xcoords, omega0, ellb,
                                                         ctxb, Vw);

    // outproj: out = outproj_w @ attn + outproj_b + out_bias
    gemm_kernel<<<ggrid, 256, 0, stream>>>(outproj_w, Vw, out,
                                           outproj_b, 0, out_bias, nullptr, /*mode=*/1);
}